// LogicLayer_52536039964873
// MI455X (gfx1250) — compile-verified
//
#include <hip/hip_runtime.h>

#define BATCH   4096
#define IN_DIM  4096
#define OUT_DIM 16384
#define BI      8           // batch rows staged in LDS per block (8*4096*4B = 128 KB)

typedef float v2f __attribute__((ext_vector_type(2)));
typedef float v4f __attribute__((ext_vector_type(4)));
typedef float v8f __attribute__((ext_vector_type(8)));
typedef int   v4i __attribute__((vector_size(16)));   // matches builtin param type

#define AS1 __attribute__((address_space(1)))
#define AS3 __attribute__((address_space(3)))

// ---------------------------------------------------------------------------
// Kernel 1: gate coefficients via softmax + V_WMMA_F32_16X16X4_F32.
//   c[j][g] = sum_k softmax(W[j])[k] * ((k >> (3-g)) & 1)   (truth table T)
// Stored refactored as k0=c0, k1=c2-c0, k2=c1-c0, k3=c0-c1-c2+c3 so the main
// kernel computes out = fma(a, fma(k3,b,k1), fma(k2,b,k0)).
// One wave = 16 gate rows; K=16 accumulated as 4 WMMA chunks of K=4.
// A 16x4 f32 layout (ISA 7.12.2): lanes 0-15 hold K={0,1}, lanes 16-31 K={2,3}.
// B 4x16 f32 layout: VGPR v, lane-half h -> row K = 2h+v, lane%16 = column N.
// ---------------------------------------------------------------------------
__global__ __launch_bounds__(256) void logic_coef_wmma(
    const float* __restrict__ W, float4* __restrict__ coef)
{
  __shared__ float cb[8][16][4];
  const int  tid  = threadIdx.x;
  const int  lane = tid & 31;
  const int  wave = tid >> 5;
  const bool lo   = lane < 16;
  const int  n    = lane & 15;
  const int  row  = blockIdx.x * 128 + wave * 16 + n;   // 16 gate rows per wave

  // Load the 16 logits for this gate row (both lane halves redundantly).
  const float4* wr = (const float4*)(W + (size_t)row * 16);
  float4 q0 = wr[0], q1 = wr[1], q2 = wr[2], q3 = wr[3];
  float w[16] = {q0.x,q0.y,q0.z,q0.w, q1.x,q1.y,q1.z,q1.w,
                 q2.x,q2.y,q2.z,q2.w, q3.x,q3.y,q3.z,q3.w};
  float m = w[0];
  #pragma unroll
  for (int k = 1; k < 16; ++k) m = fmaxf(m, w[k]);
  float e[16]; float s = 0.f;
  #pragma unroll
  for (int k = 0; k < 16; ++k) { e[k] = __expf(w[k] - m); s += e[k]; }
  const float inv = 1.f / s;

  v8f acc = {};
  #pragma unroll
  for (int c = 0; c < 4; ++c) {
    v2f A, B;
    A.x = (lo ? e[4*c + 0] : e[4*c + 2]) * inv;
    A.y = (lo ? e[4*c + 1] : e[4*c + 3]) * inv;
    const int kb = 4*c + (lo ? 0 : 2);
    B.x = (n < 4) ? (float)((kb       >> (3 - n)) & 1) : 0.f;
    B.y = (n < 4) ? (float)(((kb + 1) >> (3 - n)) & 1) : 0.f;
    // 8 args: (neg_a, A, neg_b, B, c_mod, C, reuse_a, reuse_b)
    acc = __builtin_amdgcn_wmma_f32_16x16x4_f32(false, A, false, B,
                                                (short)0, acc, false, false);
  }

  // D layout: lane n (col), half h, VGPR v -> row M = v + 8h. Columns 0..3 live.
  if (n < 4) {
    #pragma unroll
    for (int v = 0; v < 8; ++v) cb[wave][v + (lo ? 0 : 8)][n] = acc[v];
  }
  __syncthreads();
  if (lo) {
    const float c0 = cb[wave][n][0], c1 = cb[wave][n][1];
    const float c2 = cb[wave][n][2], c3 = cb[wave][n][3];
    coef[blockIdx.x * 128 + wave * 16 + n] =
        make_float4(c0, c2 - c0, c1 - c0, c0 - c1 - c2 + c3);
  }
}

// ---------------------------------------------------------------------------
// Kernel 2: bandwidth-bound gather + blend.
// Stage BI full rows of x into LDS (CDNA5: 320 KB/WGP, we use 128 KB -> 2
// blocks/WGP) via GLOBAL_LOAD_ASYNC_TO_LDS_B128 (ASYNCcnt path) when the
// builtin is available; otherwise fall back to load+ds_store staging.
// Random gathers become ds_load_b32 from LDS; x is read from HBM exactly
// once; the 256 MB output goes out via non-temporal float4 stores so it does
// not evict x/coefs from the 192 MB L2.
// ---------------------------------------------------------------------------
__global__ __launch_bounds__(256) void logic_main(
    const float* __restrict__ x, const float4* __restrict__ coef,
    const int* __restrict__ idx_a, const int* __restrict__ idx_b,
    float* __restrict__ out)
{
  __shared__ float xs[BI * IN_DIM];
  const int tid = threadIdx.x;
  const int r0  = blockIdx.x * BI;

  // Cooperative staging: 8192 float4s, 32 per thread, fully coalesced.
  const float4* src = (const float4*)(x + (size_t)r0 * IN_DIM);
  float4* dst = (float4*)xs;
#if defined(__gfx1250__) && __has_builtin(__builtin_amdgcn_global_load_async_to_lds_b128)
  #pragma unroll
  for (int i = 0; i < (BI * IN_DIM / 4) / 256; ++i)
    __builtin_amdgcn_global_load_async_to_lds_b128(
        (AS1 v4i*)(src + tid + i * 256),
        (AS3 v4i*)(dst + tid + i * 256),
        /*imm offset=*/0, /*cpol=*/0);
  #if __has_builtin(__builtin_amdgcn_s_wait_asynccnt)
  __builtin_amdgcn_s_wait_asynccnt(0);
  #else
  asm volatile("s_wait_asynccnt 0x0" ::: "memory");
  #endif
#else
  #pragma unroll
  for (int i = 0; i < (BI * IN_DIM / 4) / 256; ++i)
    dst[tid + i * 256] = src[tid + i * 256];
#endif
  __syncthreads();

  const int4* ia4 = (const int4*)idx_a;
  const int4* ib4 = (const int4*)idx_b;
  for (int chunk = 0; chunk < OUT_DIM / 1024; ++chunk) {
    const int j0 = chunk * 1024 + tid * 4;          // 4 consecutive gates/thread
    int4 pa = ia4[j0 >> 2];
    int4 pb = ib4[j0 >> 2];
    pa.x &= IN_DIM - 1; pa.y &= IN_DIM - 1; pa.z &= IN_DIM - 1; pa.w &= IN_DIM - 1;
    pb.x &= IN_DIM - 1; pb.y &= IN_DIM - 1; pb.z &= IN_DIM - 1; pb.w &= IN_DIM - 1;
    const float4 c0 = coef[j0 + 0], c1 = coef[j0 + 1];
    const float4 c2 = coef[j0 + 2], c3 = coef[j0 + 3];
    #pragma unroll
    for (int r = 0; r < BI; ++r) {
      const float* rowp = xs + r * IN_DIM;
      const float a0 = rowp[pa.x], b0 = rowp[pb.x];
      const float a1 = rowp[pa.y], b1 = rowp[pb.y];
      const float a2 = rowp[pa.z], b2 = rowp[pb.z];
      const float a3 = rowp[pa.w], b3 = rowp[pb.w];
      v4f v;
      v.x = fmaf(a0, fmaf(c0.w, b0, c0.y), fmaf(c0.z, b0, c0.x));
      v.y = fmaf(a1, fmaf(c1.w, b1, c1.y), fmaf(c1.z, b1, c1.x));
      v.z = fmaf(a2, fmaf(c2.w, b2, c2.y), fmaf(c2.z, b2, c2.x));
      v.w = fmaf(a3, fmaf(c3.w, b3, c3.y), fmaf(c3.z, b3, c3.x));
      __builtin_nontemporal_store(v, (v4f*)(out + (size_t)(r0 + r) * OUT_DIM + j0));
    }
  }
}

// ---------------------------------------------------------------------------
extern "C" void kernel_launch(void* const* d_in, const int* in_sizes, int n_in,
                              void* d_out, int out_size, void* d_ws, size_t ws_size,
                              hipStream_t stream) {
  (void)in_sizes; (void)n_in; (void)out_size; (void)ws_size;
  const float* x  = (const float*)d_in[0];
  const float* W  = (const float*)d_in[1];
  const int*   ia = (const int*)d_in[2];
  const int*   ib = (const int*)d_in[3];
  float*  out  = (float*)d_out;
  float4* coef = (float4*)d_ws;                 // 16384 * 16 B = 256 KB scratch

  logic_coef_wmma<<<OUT_DIM / 128, 256, 0, stream>>>(W, coef);
  logic_main<<<BATCH / BI, 256, 0, stream>>>(x, coef, ia, ib, out);
}